// HarmonicNoiseOscillator_46901042872745
// MI455X (gfx1250) — compile-verified
//
#include <hip/hip_runtime.h>
#include <hip/hip_bf16.h>
#include <math.h>

// ---------------------------------------------------------------------------
// HarmonicNoiseOscillator for MI455X (gfx1250, wave32)
//
// Math simplifications vs. reference:
//   P[t]   = cumsum(f0_up)/SR            (single scan; harmonic h uses frac((h+1)*P))
//   out[t] = tanh( vm^2 * SUM_h w_h*sin_h*mask_h
//                  + (0.2*vm + 0.333*(1-vm)) * S * noise[t] ),  S = SUM_h w_h
// Block phase offsets are reduced mod 1.0 (valid: harmonic multipliers are
// integers), keeping fp32 frac() precise over the 131072-sample stream.
//
// Kernel 1: per-batch block sums of f0_up/SR (blocks of 256 samples) + LDS
//           Hillis-Steele scan over the 512 blocks -> frac-reduced offsets.
// Kernel 2: per-(batch,block) workgroup of 256 threads: interp, wave32 shuffle
//           scan + cross-wave LDS fixup, then per-wave 16-timestep tiles fed
//           into chained V_WMMA_F32_16X16X4_F32 for the harmonic reduction.
// ---------------------------------------------------------------------------

#define SR        22050.0f
#define FMASK_LIM 5512.5f      // SAMPLE_RATE/4
#define T_TOTAL   131072       // 512 * 256
#define L_FRAMES  512
#define NBLOCKS   512          // T / 256
#define BLK       256

typedef __attribute__((ext_vector_type(2))) float v2f;
typedef __attribute__((ext_vector_type(8))) float v8f;

__device__ __forceinline__ float lin_interp(const float* row, int t) {
    float pos = ((float)t + 0.5f) * (1.0f / 256.0f) - 0.5f;
    pos = fminf(fmaxf(pos, 0.0f), 511.0f);
    int   i0 = (int)floorf(pos);
    int   i1 = min(i0 + 1, 511);
    float fw = pos - (float)i0;
    return row[i0] * (1.0f - fw) + row[i1] * fw;
}

__device__ __forceinline__ float masked_sine(float P, float f0u, float mult) {
    float fs = f0u * mult;
    float x  = P * mult;
    x = x - floorf(x);                       // frac -> [0,1)
    float s = __sinf(6.283185307179586f * x);
    return (fs < FMASK_LIM) ? s : 0.0f;
}

// ---------------------------------------------------------------------------
// Kernel 1: block sums + scan of phase increments. grid=(16), block=(512)
// ---------------------------------------------------------------------------
__global__ void hno_block_scan(const float* __restrict__ f0,
                               float* __restrict__ blockOff) {
    const int n = blockIdx.x;            // batch
    const int b = threadIdx.x;           // block index 0..511

    __shared__ float sF0[L_FRAMES];
    __shared__ float sS[NBLOCKS];

    sF0[b] = f0[n * L_FRAMES + b];
    __syncthreads();

    // Sum of f0_up over this 256-sample block (f0 frames served from LDS).
    float sum = 0.0f;
    const int tbase = b * BLK;
    for (int i = 0; i < BLK; ++i) {
        int   t   = tbase + i;
        float pos = ((float)t + 0.5f) * (1.0f / 256.0f) - 0.5f;
        pos = fminf(fmaxf(pos, 0.0f), 511.0f);
        int   i0 = (int)floorf(pos);
        int   i1 = min(i0 + 1, 511);
        float fw = pos - (float)i0;
        sum += sF0[i0] * (1.0f - fw) + sF0[i1] * fw;
    }
    sum *= (1.0f / SR);

    sS[b] = sum;
    __syncthreads();

    // Hillis-Steele inclusive scan over 512 block sums.
    for (int d = 1; d < NBLOCKS; d <<= 1) {
        float add = (b >= d) ? sS[b - d] : 0.0f;
        __syncthreads();
        sS[b] += add;
        __syncthreads();
    }

    float excl = sS[b] - sum;                       // exclusive prefix
    blockOff[n * NBLOCKS + b] = excl - floorf(excl); // mod 1.0 (integer harmonics)
}

// ---------------------------------------------------------------------------
// Kernel 2: main oscillator. grid=(512,16), block=(256)  (8 wave32 waves)
// ---------------------------------------------------------------------------
__global__ void __launch_bounds__(256)
hno_main(const float* __restrict__ f0, const float* __restrict__ uv,
         const float* __restrict__ wgt, const float* __restrict__ noise,
         const float* __restrict__ blockOff, float* __restrict__ out) {
    const int b    = blockIdx.x;                 // sample block 0..511
    const int n    = blockIdx.y;                 // batch 0..15
    const int tid  = threadIdx.x;                // 0..255
    const int lane = tid & 31;
    const int wave = tid >> 5;
    const int t    = b * BLK + tid;              // global sample index

    __shared__ float sP[BLK], sF[BLK], sVM[BLK], sNZ[BLK], sWS[8];

    // Pull the noise line toward the caches early.
    __builtin_prefetch(noise + (size_t)n * T_TOTAL + t, 0, 0);

    // Normalized harmonic weights (tiny; computed redundantly per thread).
    float w[8];
    float ss = 0.0f;
#pragma unroll
    for (int h = 0; h < 8; ++h) { w[h] = __expf(wgt[h]); ss += w[h] * w[h]; }
    float inv = rsqrtf(fmaxf(ss, 1e-24f));
    float S = 0.0f;
#pragma unroll
    for (int h = 0; h < 8; ++h) { w[h] *= inv; S += w[h]; }

    // Frame-rate -> sample-rate linear interpolation.
    const float f0u = lin_interp(f0 + n * L_FRAMES, t);
    const float vm  = lin_interp(uv + n * L_FRAMES, t);
    const float nz  = noise[(size_t)n * T_TOTAL + t];

    // Inclusive scan of phase increments: wave32 shuffle scan + LDS fixup.
    float v = f0u * (1.0f / SR);
#pragma unroll
    for (int d = 1; d < 32; d <<= 1) {
        float u = __shfl_up(v, d, 32);
        if (lane >= d) v += u;
    }
    if (lane == 31) sWS[wave] = v;
    __syncthreads();
    float woff = 0.0f;
    for (int wv = 0; wv < wave; ++wv) woff += sWS[wv];
    const float P = blockOff[n * NBLOCKS + b] + woff + v;

    sP[tid] = P; sF[tid] = f0u; sVM[tid] = vm; sNZ[tid] = nz;
    __syncthreads();

    // Per-wave: two 16-timestep tiles. Harmonic reduction via chained
    // V_WMMA_F32_16X16X4_F32:
    //   B (4x16): K = harmonic (4 of 8), N = timestep; each lane computes 2
    //             masked sines per WMMA (lanes 0-15: K=0,1; lanes 16-31: K=2,3)
    //   A (16x4): normalized weights broadcast down the 16 rows
    //   D row M=0, lanes 0-15  ==  SUM_h w_h*sin_h*mask_h for the 16 timesteps
#pragma unroll
    for (int tile = 0; tile < 2; ++tile) {
        const int   lb  = wave * 32 + tile * 16;     // tile base (local)
        const int   src = lb + (lane & 15);
        const float Pt  = sP[src];
        const float Ft  = sF[src];
        const int   kh  = (lane < 16) ? 0 : 2;       // K-slot offset this half-wave

        v8f acc = {};
#pragma unroll
        for (int c = 0; c < 2; ++c) {                // harmonic groups {1..4},{5..8}
            const int hb = 4 * c;
            v2f bm, am;
            bm.x = masked_sine(Pt, Ft, (float)(hb + kh + 1));
            bm.y = masked_sine(Pt, Ft, (float)(hb + kh + 2));
            am.x = (lane < 16) ? w[hb + 0] : w[hb + 2];
            am.y = (lane < 16) ? w[hb + 1] : w[hb + 3];
            acc = __builtin_amdgcn_wmma_f32_16x16x4_f32(
                /*neg_a=*/false, am, /*neg_b=*/false, bm,
                /*c_mod=*/(short)0, acc, /*reuse_a=*/false, /*reuse_b=*/false);
        }

        if (lane < 16) {                             // D: VGPR0, M=0, N=lane
            const int   li  = lb + lane;
            const float vmv = sVM[li];
            const float nzv = sNZ[li];
            const float g   = vmv * 0.2f + (1.0f - vmv) * 0.333f;
            const float val = tanhf(vmv * vmv * acc[0] + g * S * nzv);
            out[(size_t)n * T_TOTAL + b * BLK + li] = val;
        }
    }
}

// ---------------------------------------------------------------------------
extern "C" void kernel_launch(void* const* d_in, const int* in_sizes, int n_in,
                              void* d_out, int out_size, void* d_ws, size_t ws_size,
                              hipStream_t stream) {
    const float* f0    = (const float*)d_in[0];   // [16,1,512]
    const float* uv    = (const float*)d_in[1];   // [16,1,512]
    const float* wgt   = (const float*)d_in[2];   // [1,8,1]
    const float* noise = (const float*)d_in[3];   // [16,1,131072]
    float*       out   = (float*)d_out;           // [16,1,131072]
    float*       boff  = (float*)d_ws;            // 16*512 floats = 32 KB scratch

    hipLaunchKernelGGL(hno_block_scan, dim3(16), dim3(NBLOCKS), 0, stream, f0, boff);
    hipLaunchKernelGGL(hno_main, dim3(NBLOCKS, 16), dim3(BLK), 0, stream,
                       f0, uv, wgt, noise, boff, out);
}